// SplineEmbedding_61907658605068
// MI455X (gfx1250) — compile-verified
//
#include <hip/hip_runtime.h>
#include <hip/hip_bf16.h>

#define ACTIONS 64
#define EMB 32
#define DELTA 16
#define BN_EPS 1e-5f
#define ROWS_PER_STAT_BLOCK 128   // 8 waves: 4 col-groups x 2 row-halves of 64 rows

typedef float v2f __attribute__((ext_vector_type(2)));
typedef float v4f __attribute__((ext_vector_type(4)));
typedef float v8f __attribute__((ext_vector_type(8)));

// ---------------------------------------------------------------------------
// Stage 1: per-block partial column sums / sums-of-squares via V_WMMA_F32_16X16X4_F32
//   B tile (4x16 f32): VGPR0 = {K=0 | lanes 0-15, K=2 | lanes 16-31}, VGPR1 = {K=1, K=3}
//   A tile (16x4 f32): VGPR0 = {K=0 | M=lane0-15, K=2 | M=lane16-31}, VGPR1 = {K=1, K=3}
//   => a tile loaded in B layout is ALSO its own transpose in A layout, so
//      wmma(tile, tile) accumulates B^T B whose diagonal is sum(x^2) per column.
// Handles only full 128-row blocks: loads are unguarded and fully unrolled so
// each becomes global_load_b32 with an immediate offset off one base address.
// Tail rows (n % 128) are folded in by the finalize kernel.
// ---------------------------------------------------------------------------
__global__ __launch_bounds__(256)
void spline_stats_kernel(const float* __restrict__ x, float* __restrict__ partials) {
    __shared__ float sq_tile[8 * 256];   // one 16x16 f32 tile per wave
    __shared__ float blk_sum[ACTIONS];
    __shared__ float blk_sq[ACTIONS];

    const int tid  = threadIdx.x;
    const int wave = tid >> 5;
    const int lane = tid & 31;
    if (tid < ACTIONS) { blk_sum[tid] = 0.0f; blk_sq[tid] = 0.0f; }
    __syncthreads();

    const int cg   = wave & 3;          // column group (16 columns each)
    const int half = wave >> 2;         // which 64-row half of the block's 128 rows
    const int col  = cg * 16 + (lane & 15);
    const int rsel = lane >> 4;         // 0 -> K=0/1, 1 -> K=2/3
    const int base_row = blockIdx.x * ROWS_PER_STAT_BLOCK + half * 64;

    // One base address per lane; every load below is an immediate offset off it.
    const float* __restrict__ p = x + (size_t)(base_row + 2 * rsel) * ACTIONS + col;

    v8f acc_s = {};
    v8f acc_q = {};
    v2f ones; ones[0] = 1.0f; ones[1] = 1.0f;

#pragma unroll
    for (int step = 0; step < 16; ++step) {
        v2f b;
        b[0] = p[step * 4 * ACTIONS];            // K = 2*rsel   row
        b[1] = p[step * 4 * ACTIONS + ACTIONS];  // K = 2*rsel+1 row
        // column sums: every output row m holds sum_k B[k][n]
        acc_s = __builtin_amdgcn_wmma_f32_16x16x4_f32(false, ones, false, b,
                                                      (short)0, acc_s, false, false);
        // B^T B: diagonal holds sum_k B[k][n]^2
        acc_q = __builtin_amdgcn_wmma_f32_16x16x4_f32(false, b, false, b,
                                                      (short)0, acc_q, false, false);
    }

    // Spill sq accumulator tile to LDS to pull out the diagonal.
    // C/D layout: lanes 0-15 -> N=lane, M = v (+8 for lanes 16-31).
    {
        float* wt = &sq_tile[wave * 256];
        const int nn = lane & 15;
        const int mb = (lane >> 4) * 8;
#pragma unroll
        for (int v = 0; v < 8; ++v) wt[(mb + v) * 16 + nn] = acc_q[v];
    }
    __syncthreads();

    if (lane < 16) {
        const float psum = acc_s[0];                         // row M=0 of sum tile
        const float psq  = sq_tile[wave * 256 + lane * 17];  // diagonal element
        atomicAdd(&blk_sum[cg * 16 + lane], psum);           // 2 commutative adds -> deterministic
        atomicAdd(&blk_sq [cg * 16 + lane], psq);
    }
    __syncthreads();

    if (tid < ACTIONS)            partials[(size_t)blockIdx.x * 128 + tid] = blk_sum[tid];
    else if (tid < 2 * ACTIONS)   partials[(size_t)blockIdx.x * 128 + tid] = blk_sq[tid - ACTIONS];
}

// ---------------------------------------------------------------------------
// Stage 2: fixed-order reduction of partials (+ any tail rows not covered by
// full 128-row stat blocks) -> per-column scale/shift
// ---------------------------------------------------------------------------
__global__ __launch_bounds__(64)
void spline_finalize_kernel(const float* __restrict__ x,
                            const float* __restrict__ partials,
                            const float* __restrict__ bn_weight,
                            const float* __restrict__ bn_bias,
                            float* __restrict__ params, int n, int nblocks) {
    const int c = threadIdx.x;
    if (c >= ACTIONS) return;
    float s = 0.0f, q = 0.0f;
    for (int b = 0; b < nblocks; ++b) {
        s += partials[(size_t)b * 128 + c];
        q += partials[(size_t)b * 128 + ACTIONS + c];
    }
    for (int r = nblocks * ROWS_PER_STAT_BLOCK; r < n; ++r) {   // tail (none when n%128==0)
        const float v = x[(size_t)r * ACTIONS + c];
        s += v; q += v * v;
    }
    const float inv_n = 1.0f / (float)n;
    const float mean  = s * inv_n;
    const float var   = q * inv_n - mean * mean;   // biased batch variance
    const float scale = bn_weight[c] * rsqrtf(var + BN_EPS);
    const float shift = bn_bias[c] - mean * scale;
    params[c]           = scale;
    params[ACTIONS + c] = shift;
}

// ---------------------------------------------------------------------------
// Stage 3: streaming gather + lerp. 8 threads per (row, action) pair, one
// float4 of the 32-wide emb vector each -> a wave covers 4 pairs and writes
// 512 contiguous bytes. Non-temporal stores keep the 134 MB output stream
// from evicting emb_table/x out of L2.
// ---------------------------------------------------------------------------
__global__ __launch_bounds__(256)
void spline_gather_kernel(const float* __restrict__ x,
                          const float* __restrict__ params,
                          const float* __restrict__ emb,
                          v4f* __restrict__ out, long long npairs) {
    const long long gid  = (long long)blockIdx.x * blockDim.x + threadIdx.x;
    const long long pair = gid >> 3;
    const int       j    = (int)(gid & 7);
    if (pair >= npairs) return;

    const int act = (int)(pair & (ACTIONS - 1));
    const float xv    = x[pair];
    const float scale = params[act];
    const float shift = params[ACTIONS + act];
    const float xn    = fmaf(xv, scale, shift);

#if __has_builtin(__builtin_amdgcn_tanhf)
    float t = __builtin_amdgcn_tanhf(xn);        // gfx1250 v_tanh_f32
#else
    const float ax = fabsf(xn);
    const float e  = __expf(-2.0f * ax);
    float t = copysignf((1.0f - e) / (1.0f + e), xn);
#endif
    t = fminf(fmaxf(t, -1.0f + 1e-5f), 1.0f - 1e-5f);

    const float s   = t * (float)DELTA;
    const float xlf = floorf(s);
    const float xhf = floorf(s + 1.0f);
    const int   xli = ACTIONS * ((int)xlf + DELTA) + act;
    const int   xhi = ACTIONS * ((int)xhf + DELTA) + act;
    const float wh  = (float)DELTA * (t - xlf * (1.0f / (float)DELTA));
    const float wl  = (float)DELTA * (xhf * (1.0f / (float)DELTA) - t);

    const v4f* __restrict__ embv = (const v4f*)emb;
    const v4f bl = embv[(long long)xli * (EMB / 4) + j];
    const v4f bh = embv[(long long)xhi * (EMB / 4) + j];

    v4f o;
    o[0] = bh[0] * wh + bl[0] * wl;
    o[1] = bh[1] * wh + bl[1] * wl;
    o[2] = bh[2] * wh + bl[2] * wl;
    o[3] = bh[3] * wh + bl[3] * wl;
    __builtin_nontemporal_store(o, &out[pair * (EMB / 4) + j]);
}

// ---------------------------------------------------------------------------
extern "C" void kernel_launch(void* const* d_in, const int* in_sizes, int n_in,
                              void* d_out, int out_size, void* d_ws, size_t ws_size,
                              hipStream_t stream) {
    const float* x   = (const float*)d_in[0];
    const float* bnw = (const float*)d_in[1];
    const float* bnb = (const float*)d_in[2];
    const float* emb = (const float*)d_in[3];

    const int n = in_sizes[0] / ACTIONS;                   // 16384
    const int stat_blocks = n / ROWS_PER_STAT_BLOCK;       // full blocks only; tail in finalize

    float* partials = (float*)d_ws;                         // stat_blocks * 128 floats
    float* params   = partials + (size_t)stat_blocks * 128; // scale[64] ++ shift[64]

    if (stat_blocks > 0)
        spline_stats_kernel<<<stat_blocks, 256, 0, stream>>>(x, partials);
    spline_finalize_kernel<<<1, 64, 0, stream>>>(x, partials, bnw, bnb, params, n, stat_blocks);

    const long long npairs  = (long long)n * ACTIONS;
    const long long threads = npairs * 8;                   // 8 threads per pair
    const int blocks = (int)((threads + 255) / 256);
    spline_gather_kernel<<<blocks, 256, 0, stream>>>(x, params, emb, (v4f*)d_out, npairs);
}